// LearnedDownsampling_3556232921275
// MI455X (gfx1250) — compile-verified
//
#include <hip/hip_runtime.h>
#include <hip/hip_bf16.h>

// ---------------- problem constants ----------------
#define BATCH   16
#define NPG     62500            // nodes per graph
#define NTOT    (BATCH * NPG)    // 1,000,000
#define FEAT    128
#define KSEL    31250            // top-K per segment
#define TILE    256
#define NTILES  ((NPG + TILE - 1) / TILE)   // 245
#define SEGHIST (256 * NTILES)              // 62720 entries per segment

typedef __attribute__((ext_vector_type(2))) float v2f;
typedef __attribute__((ext_vector_type(8))) float v8f;

// Map float to u32 key such that ASCENDING key == DESCENDING score.
// orderable(f): flip sign bit for positives, invert for negatives (ascending),
// then bitwise-NOT to reverse. Stable LSD sort then matches jax.lax.top_k
// (descending, ties keep lower index first).
__device__ __forceinline__ unsigned score_to_key(float s) {
    unsigned u = __float_as_uint(s);
    u = (u & 0x80000000u) ? (~u) : (u | 0x80000000u); // ascending orderable
    return ~u;                                        // reverse -> descending
}

// ---------------- 1) scores via v_wmma_f32_16x16x4_f32 ----------------
// One wave computes 16 rows. A (16x4 f32, ISA layout): lane L<16 holds row L,
// K={0,1}; lane L>=16 holds row L-16, K={2,3}. B (4x16) uses the same per-lane
// K mapping with W replicated across all 16 columns, so every column of D
// carries the same 16 scores. Chain C<-D for 32 steps to cover K=128.
__global__ void __launch_bounds__(128)
score_kernel(const float* __restrict__ x, const float* __restrict__ W,
             const float* __restrict__ bias_p, unsigned* __restrict__ keyA) {
    const int wave = blockIdx.x * 4 + (threadIdx.x >> 5);
    const int lane = threadIdx.x & 31;
    const int r0   = wave * 16;
    const int row  = r0 + (lane & 15);
    const int kb   = (lane >> 4) * 2;           // 0 or 2
    const float* __restrict__ xr = x + (size_t)row * FEAT + kb;

    v8f acc = {0.f, 0.f, 0.f, 0.f, 0.f, 0.f, 0.f, 0.f};
#pragma unroll
    for (int t = 0; t < 32; ++t) {
        v2f a, b;
        a.x = xr[t * 4];
        a.y = xr[t * 4 + 1];
        b.x = W[t * 4 + kb];
        b.y = W[t * 4 + kb + 1];
        acc = __builtin_amdgcn_wmma_f32_16x16x4_f32(
            /*neg_a=*/false, a, /*neg_b=*/false, b,
            /*c_mod=*/(short)0, acc, /*reuse_a=*/false, /*reuse_b=*/false);
    }

    const float bias = bias_p[0];
    // D layout: VGPR v holds M=v (lanes 0-15) and M=v+8 (lanes 16-31).
    // Lanes 0 and 16 each emit 8 keys.
    if ((lane & 15) == 0) {
        const int mbase = (lane >> 4) * 8;
#pragma unroll
        for (int v = 0; v < 8; ++v) {
            keyA[r0 + mbase + v] = score_to_key(acc[v] + bias);
        }
    }
}

// ---------------- 2) segmented stable LSD radix sort ----------------
__global__ void init_val_kernel(unsigned* __restrict__ valA) {
    int i = blockIdx.x * blockDim.x + threadIdx.x;
    if (i < NTOT) valA[i] = (unsigned)(i % NPG);   // local index within segment
}

__global__ void hist_kernel(const unsigned* __restrict__ key_in,
                            unsigned* __restrict__ hist, int shift) {
    __shared__ unsigned h[256];
    const int tile = blockIdx.x % NTILES;
    const int seg  = blockIdx.x / NTILES;
    h[threadIdx.x] = 0;
    __syncthreads();
    const int i = tile * TILE + threadIdx.x;
    if (i < NPG) {
        unsigned d = (key_in[(size_t)seg * NPG + i] >> shift) & 255u;
        atomicAdd(&h[d], 1u);
    }
    __syncthreads();
    hist[((size_t)seg * 256 + threadIdx.x) * NTILES + tile] = h[threadIdx.x];
}

// One 1024-thread workgroup per segment: in-place exclusive prefix sum over
// the segment's 256*NTILES histogram entries (digit-major, then tile).
__global__ void __launch_bounds__(1024)
scan_kernel(unsigned* __restrict__ hist) {
    const int seg = blockIdx.x;
    unsigned* __restrict__ p = hist + (size_t)seg * SEGHIST;
    const int t = threadIdx.x;
    const int C = (SEGHIST + 1023) / 1024;   // 62 entries per thread

    // pass A: per-thread chunk sums
    unsigned sum = 0;
    for (int c = 0; c < C; ++c) {
        int idx = t * C + c;
        if (idx < SEGHIST) sum += p[idx];
    }
    __shared__ unsigned s[1024];
    s[t] = sum;
    __syncthreads();
    // inclusive Hillis-Steele block scan
    for (int off = 1; off < 1024; off <<= 1) {
        unsigned v = (t >= off) ? s[t - off] : 0u;
        __syncthreads();
        s[t] += v;
        __syncthreads();
    }
    unsigned running = s[t] - sum;   // exclusive base for this thread's chunk
    // pass B: rewrite chunk with exclusive prefix
    for (int c = 0; c < C; ++c) {
        int idx = t * C + c;
        if (idx < SEGHIST) {
            unsigned v = p[idx];
            p[idx] = running;
            running += v;
        }
    }
}

__global__ void scatter_kernel(const unsigned* __restrict__ key_in,
                               const unsigned* __restrict__ val_in,
                               unsigned* __restrict__ key_out,
                               unsigned* __restrict__ val_out,
                               const unsigned* __restrict__ hist, int shift) {
    __shared__ unsigned sdig[TILE];
    const int tile = blockIdx.x % NTILES;
    const int seg  = blockIdx.x / NTILES;
    const int i    = tile * TILE + threadIdx.x;
    const bool valid = (i < NPG);

    unsigned key = 0u, val = 0u, dig = 256u;   // 256 = sentinel, never matches
    if (valid) {
        key = key_in[(size_t)seg * NPG + i];
        val = val_in[(size_t)seg * NPG + i];
        dig = (key >> shift) & 255u;
    }
    sdig[threadIdx.x] = dig;
    __syncthreads();
    if (valid) {
        unsigned lr = 0;                        // stable rank within tile
        for (int j = 0; j < (int)threadIdx.x; ++j) lr += (sdig[j] == dig);
        unsigned base = hist[((size_t)seg * 256 + dig) * NTILES + tile];
        unsigned dst  = base + lr;
        key_out[(size_t)seg * NPG + dst] = key;
        val_out[(size_t)seg * NPG + dst] = val;
    }
}

// ---------------- 3) gather outputs ----------------
// One wave per output row: 32 lanes x float4 = 512B coalesced copy.
// d_out layout: x_out [B*K*F] ++ batch_out [B*K] ++ perm [B*K] (all as f32).
__global__ void __launch_bounds__(256)
gather_kernel(const float* __restrict__ x, const unsigned* __restrict__ sorted_val,
              float* __restrict__ out) {
    const int row  = (blockIdx.x * blockDim.x + threadIdx.x) >> 5;
    const int lane = threadIdx.x & 31;
    if (row >= BATCH * KSEL) return;
    const int seg  = row / KSEL;
    const int rank = row - seg * KSEL;
    const unsigned lidx = sorted_val[(size_t)seg * NPG + rank];
    const size_t g = (size_t)seg * NPG + lidx;

    const float4* __restrict__ src = (const float4*)(x + g * FEAT);
    float4* __restrict__ dst = (float4*)(out + (size_t)row * FEAT);
    dst[lane] = src[lane];

    if (lane == 0) {
        const size_t ofs = (size_t)BATCH * KSEL * FEAT;
        out[ofs + row]                       = (float)seg;   // batch_out
        out[ofs + (size_t)BATCH * KSEL + row] = (float)g;    // perm
    }
}

// ---------------- host launcher ----------------
extern "C" void kernel_launch(void* const* d_in, const int* in_sizes, int n_in,
                              void* d_out, int out_size, void* d_ws, size_t ws_size,
                              hipStream_t stream) {
    const float* x  = (const float*)d_in[0];
    // d_in[1] = batch (int32) — segments are contiguous/equal, recomputed on the fly
    const float* W  = (const float*)d_in[2];
    const float* b  = (const float*)d_in[3];
    float* out      = (float*)d_out;

    // workspace carve-up (~20 MB)
    unsigned* keyA = (unsigned*)d_ws;
    unsigned* keyB = keyA + NTOT;
    unsigned* valA = keyB + NTOT;
    unsigned* valB = valA + NTOT;
    unsigned* hist = valB + NTOT;   // BATCH * 256 * NTILES u32

    // 1) scores -> descending-order keys (WMMA f32 16x16x4)
    score_kernel<<<NTOT / 64, 128, 0, stream>>>(x, W, b, keyA);

    // 2) stable segmented LSD radix sort of (key, local index)
    init_val_kernel<<<(NTOT + 255) / 256, 256, 0, stream>>>(valA);

    const unsigned* ki[4] = {keyA, keyB, keyA, keyB};
    const unsigned* vi[4] = {valA, valB, valA, valB};
    unsigned*       ko[4] = {keyB, keyA, keyB, keyA};
    unsigned*       vo[4] = {valB, valA, valB, valA};
    for (int p = 0; p < 4; ++p) {
        const int shift = 8 * p;
        hist_kernel<<<BATCH * NTILES, 256, 0, stream>>>(ki[p], hist, shift);
        scan_kernel<<<BATCH, 1024, 0, stream>>>(hist);
        scatter_kernel<<<BATCH * NTILES, 256, 0, stream>>>(ki[p], vi[p], ko[p], vo[p],
                                                           hist, shift);
    }
    // after 4 passes the fully sorted (descending score, stable) vals are in valA

    // 3) gather x[perm], batch_out, perm
    const int rows = BATCH * KSEL;                 // 500,000
    gather_kernel<<<(rows * 32 + 255) / 256, 256, 0, stream>>>(x, valA, out);
    (void)in_sizes; (void)n_in; (void)out_size; (void)ws_size;
}